// FuzzyInferenceLayer_39273180954962
// MI455X (gfx1250) — compile-verified
//
#include <hip/hip_runtime.h>

#define N_TERMS 5
#define N_MF    8
#define N_HALF  625      /* 5^4 */
#define N_RULES 390625   /* 625*625 */
#define NPAD    640      /* padded to tile multiple */
#define NT_IN   39       /* interior tiles per dim: 39*16 = 624 */
#define N_BORDER 1249    /* row 624 (625) + col 624 rows 0..623 (624) */
#define SPLIT   16       /* blocks per batch row */
#define EPS_F   1e-12f

typedef __attribute__((ext_vector_type(2)))  float v2f;
typedef __attribute__((ext_vector_type(8)))  float v8f;

__global__ __launch_bounds__(256)
void fuzzy_outer_wmma(const float* __restrict__ x, float* __restrict__ out)
{
    __shared__ float xb[N_TERMS * N_MF];   // 40 floats of x for this batch row
    __shared__ float A[NPAD];              // products of MFs 0..3 (scaled), zero-padded
    __shared__ float B[NPAD];              // products of MFs 4..7, zero-padded
    __shared__ float red[256];

    const int b   = blockIdx.x;
    const int tid = threadIdx.x;

    if (tid < N_TERMS * N_MF)
        xb[tid] = x[b * (N_TERMS * N_MF) + tid];
    __syncthreads();

    // ---- Stage 1: factor products A[i], B[j] and their abs-sums ----
    float sa = 0.f, sb = 0.f;
    for (int i = tid; i < NPAD; i += 256) {
        float av = 0.f, bv = 0.f;
        if (i < N_HALF) {
            int d0 = i / 125;
            int r  = i - d0 * 125;
            int d1 = r / 25;  r -= d1 * 25;
            int d2 = r / 5;
            int d3 = r - d2 * 5;
            av = xb[d0 * 8 + 0] * xb[d1 * 8 + 1] * xb[d2 * 8 + 2] * xb[d3 * 8 + 3];
            bv = xb[d0 * 8 + 4] * xb[d1 * 8 + 5] * xb[d2 * 8 + 6] * xb[d3 * 8 + 7];
        }
        A[i] = av;  B[i] = bv;
        sa += fabsf(av);  sb += fabsf(bv);
    }

    // deterministic tree reductions (no float atomics)
    red[tid] = sa;  __syncthreads();
    for (int s = 128; s > 0; s >>= 1) {
        if (tid < s) red[tid] += red[tid + s];
        __syncthreads();
    }
    const float sumA = red[0];
    __syncthreads();
    red[tid] = sb;  __syncthreads();
    for (int s = 128; s > 0; s >>= 1) {
        if (tid < s) red[tid] += red[tid + s];
        __syncthreads();
    }
    const float sumB = red[0];
    __syncthreads();

    const float denom = fmaxf(sumA * sumB, EPS_F);
    const float inv   = 1.0f / denom;

    // fold normalizer into A
    for (int i = tid; i < NPAD; i += 256) A[i] *= inv;
    __syncthreads();

    float* outb = out + (size_t)b * N_RULES;

    // ---- Border (row 624 + col 624): plain VALU, done by y==0 blocks ----
    if (blockIdx.y == 0) {
        const float a624 = A[624];
        const float b624 = B[624];
        for (int k = tid; k < N_BORDER; k += 256) {
            int row, col;
            float w;
            if (k < N_HALF) { row = 624; col = k;         w = a624 * B[col]; }
            else            { row = k - N_HALF; col = 624; w = A[row] * b624; }
            outb[(size_t)row * N_HALF + col] = w;
        }
    }

    // ---- Stage 2: interior 16x16 outer-product tiles via WMMA (no predication) ----
    const int lane   = tid & 31;
    const int wave   = tid >> 5;                 // 0..7
    const int gwave  = blockIdx.y * 8 + wave;    // 0..127
    const int lane15 = lane & 15;
    const bool loHalf = (lane < 16);
    const int colOff  = lane15;
    const int rowOff  = (lane >> 4) << 3;        // 0 or 8

    for (int t = gwave; t < NT_IN * NT_IN; t += 8 * SPLIT) {
        const int ti = t / NT_IN;                // wave-uniform -> SALU
        const int tj = t - ti * NT_IN;
        const int i0 = ti * 16, j0 = tj * 16;

        // rank-1 tile: only the K=0 slice (lanes 0..15, first VGPR) is populated.
        // Unconditional in-bounds LDS loads, value-masked (cndmask, no exec churn).
        const float aRaw = A[i0 + lane15];
        const float bRaw = B[j0 + lane15];
        const float aval = loHalf ? aRaw : 0.0f;
        const float bval = loHalf ? bRaw : 0.0f;

        v2f av = { aval, 0.0f };
        v2f bv = { bval, 0.0f };
        v8f c = {};
        c = __builtin_amdgcn_wmma_f32_16x16x4_f32(false, av, false, bv,
                                                  (short)0, c, false, false);

        // C layout: VGPR v -> (M = v + 8*(lane>=16), N = lane%16)
        // One per-lane address, 8 stores with immediate offsets (v*625*4 bytes).
        float* p = outb + ((i0 + rowOff) * N_HALF + j0 + colOff);
        #pragma unroll
        for (int v = 0; v < 8; ++v)
            p[v * N_HALF] = c[v];
    }
}

extern "C" void kernel_launch(void* const* d_in, const int* in_sizes, int n_in,
                              void* d_out, int out_size, void* d_ws, size_t ws_size,
                              hipStream_t stream)
{
    (void)n_in; (void)out_size; (void)d_ws; (void)ws_size;
    const float* x = (const float*)d_in[0];
    // d_in[1] (combos) is intentionally unused: digits are recomputed analytically.
    float* out = (float*)d_out;

    const int batches = in_sizes[0] / (N_TERMS * N_MF);   // 64
    dim3 grid(batches, SPLIT);
    fuzzy_outer_wmma<<<grid, 256, 0, stream>>>(x, out);
}